// ProteinStructureLoss_9096740733747
// MI455X (gfx1250) — compile-verified
//
#include <hip/hip_runtime.h>
#include <math.h>

typedef float v2f __attribute__((ext_vector_type(2)));
typedef float v8f __attribute__((ext_vector_type(8)));

#define LSEQ 1024
#define NB   16
#define EPSF 1e-6f

__device__ __forceinline__ float fast_sqrt(float x) {
    return __builtin_amdgcn_sqrtf(x);
}
__device__ __forceinline__ float fast_rcp(float x) {
    return __builtin_amdgcn_rcpf(x);
}

// ---------------------------------------------------------------------------
// init: zero the 4 output scalars and the per-batch dist/clash accumulators
// ---------------------------------------------------------------------------
__global__ void k_init(float* __restrict__ out, float* __restrict__ acc) {
    int t = threadIdx.x;
    if (t < 4)  out[t] = 0.0f;
    if (t < 2 * NB) acc[t] = 0.0f;
}

// ---------------------------------------------------------------------------
// compact: stable-partition valid residues to the front (matches stable
// argsort of !mask), store float4 (x,y,z,|x|^2), zero padding, record nvalid
// ---------------------------------------------------------------------------
__global__ __launch_bounds__(LSEQ)
void k_compact(const float* __restrict__ pred, const float* __restrict__ target,
               const int* __restrict__ gmask,
               float4* __restrict__ pc, float4* __restrict__ tc,
               int* __restrict__ nvalid) {
    __shared__ int sbuf[LSEQ];
    const int b = blockIdx.x;
    const int t = threadIdx.x;
    const int g = (gmask[b * LSEQ + t] != 0) ? 1 : 0;
    sbuf[t] = g;
    __syncthreads();
    // Hillis-Steele inclusive scan over the block
    for (int off = 1; off < LSEQ; off <<= 1) {
        int v = sbuf[t];
        int a = (t >= off) ? sbuf[t - off] : 0;
        __syncthreads();
        sbuf[t] = v + a;
        __syncthreads();
    }
    const int incl = sbuf[t];
    const int nv   = sbuf[LSEQ - 1];
    // zero fill first, then scatter the valid entries
    pc[b * LSEQ + t] = make_float4(0.f, 0.f, 0.f, 0.f);
    tc[b * LSEQ + t] = make_float4(0.f, 0.f, 0.f, 0.f);
    __syncthreads();
    if (g) {
        const int dst = incl - 1;
        const int src = (b * LSEQ + t) * 3;
        float px = pred[src + 0], py = pred[src + 1], pz = pred[src + 2];
        float tx = target[src + 0], ty = target[src + 1], tz = target[src + 2];
        pc[b * LSEQ + dst] = make_float4(px, py, pz, px * px + py * py + pz * pz);
        tc[b * LSEQ + dst] = make_float4(tx, ty, tz, tx * tx + ty * ty + tz * tz);
    }
    if (t == 0) nvalid[b] = nv;
}

// ---------------------------------------------------------------------------
// angle + torsion losses (sin/cos packed without transcendentals:
//   sin(arccos(c)) = sqrt(1-c^2),  cos(sign*a) = cos(a), sin(sign*a)=sign*sin(a))
// ---------------------------------------------------------------------------
__device__ __forceinline__ float dot3f(float ax, float ay, float az,
                                       float bx, float by, float bz) {
    return ax * bx + ay * by + az * bz;
}

__global__ __launch_bounds__(256)
void k_angle(const float4* __restrict__ pc, const float4* __restrict__ tc,
             const int* __restrict__ nvalid, float* __restrict__ out) {
    const int b = blockIdx.x;
    const int t = threadIdx.x;
    const int nv = nvalid[b];
    const float4* P = pc + b * LSEQ;
    const float4* T = tc + b * LSEQ;

    float sumA = 0.f, cntA = 0.f, sumT = 0.f, cntT = 0.f;

    // ---- bond angles: term j uses c[j..j+2], weight maskf[j+1] ----
    for (int j = t; j < LSEQ - 2; j += 256) {
        if (j + 1 < nv) {
            cntA += 1.f;
            float4 a0 = P[j], a1 = P[j + 1], a2 = P[j + 2];
            float v1x = a0.x - a1.x, v1y = a0.y - a1.y, v1z = a0.z - a1.z;
            float v2x = a2.x - a1.x, v2y = a2.y - a1.y, v2z = a2.z - a1.z;
            float cp = dot3f(v1x, v1y, v1z, v2x, v2y, v2z) *
                       fast_rcp(fast_sqrt(dot3f(v1x, v1y, v1z, v1x, v1y, v1z)) *
                                fast_sqrt(dot3f(v2x, v2y, v2z, v2x, v2y, v2z)) + EPSF);
            cp = fminf(1.f, fmaxf(-1.f, cp));
            float sp = fast_sqrt(fmaxf(1.f - cp * cp, 0.f));

            float4 b0 = T[j], b1 = T[j + 1], b2 = T[j + 2];
            float w1x = b0.x - b1.x, w1y = b0.y - b1.y, w1z = b0.z - b1.z;
            float w2x = b2.x - b1.x, w2y = b2.y - b1.y, w2z = b2.z - b1.z;
            float ct = dot3f(w1x, w1y, w1z, w2x, w2y, w2z) *
                       fast_rcp(fast_sqrt(dot3f(w1x, w1y, w1z, w1x, w1y, w1z)) *
                                fast_sqrt(dot3f(w2x, w2y, w2z, w2x, w2y, w2z)) + EPSF);
            ct = fminf(1.f, fmaxf(-1.f, ct));
            float st = fast_sqrt(fmaxf(1.f - ct * ct, 0.f));

            float ds = sp - st, dc = cp - ct;
            sumA += ds * ds + dc * dc;
        }
    }

    // ---- dihedrals: term j uses c[j..j+3], weight maskf[j+2] ----
    for (int j = t; j < LSEQ - 3; j += 256) {
        if (j + 2 < nv) {
            cntT += 1.f;
            const float4* C[2] = { P, T };
            float sv[2], cv[2];
            #pragma unroll
            for (int s = 0; s < 2; ++s) {
                float4 q0 = C[s][j], q1 = C[s][j + 1], q2 = C[s][j + 2], q3 = C[s][j + 3];
                float v1x = q1.x - q0.x, v1y = q1.y - q0.y, v1z = q1.z - q0.z;
                float v2x = q2.x - q1.x, v2y = q2.y - q1.y, v2z = q2.z - q1.z;
                float v3x = q3.x - q2.x, v3y = q3.y - q2.y, v3z = q3.z - q2.z;
                float n1x = v1y * v2z - v1z * v2y;
                float n1y = v1z * v2x - v1x * v2z;
                float n1z = v1x * v2y - v1y * v2x;
                float n2x = v2y * v3z - v2z * v3y;
                float n2y = v2z * v3x - v2x * v3z;
                float n2z = v2x * v3y - v2y * v3x;
                float cp = dot3f(n1x, n1y, n1z, n2x, n2y, n2z) *
                           fast_rcp(fast_sqrt(dot3f(n1x, n1y, n1z, n1x, n1y, n1z)) *
                                    fast_sqrt(dot3f(n2x, n2y, n2z, n2x, n2y, n2z)) + EPSF);
                cp = fminf(1.f, fmaxf(-1.f, cp));
                // sign(dot(cross(n1,n2), v2))
                float cxx = n1y * n2z - n1z * n2y;
                float cxy = n1z * n2x - n1x * n2z;
                float cxz = n1x * n2y - n1y * n2x;
                float sg = dot3f(cxx, cxy, cxz, v2x, v2y, v2z);
                float sgn = (sg > 0.f ? 1.f : 0.f) - (sg < 0.f ? 1.f : 0.f);
                sv[s] = sgn * fast_sqrt(fmaxf(1.f - cp * cp, 0.f));
                cv[s] = cp;
            }
            float ds = sv[0] - sv[1], dc = cv[0] - cv[1];
            sumT += ds * ds + dc * dc;
        }
    }

    // block reduction of (sumA, cntA, sumT, cntT)
    __shared__ float red[256][4];
    red[t][0] = sumA; red[t][1] = cntA; red[t][2] = sumT; red[t][3] = cntT;
    __syncthreads();
    for (int s = 128; s > 0; s >>= 1) {
        if (t < s) {
            red[t][0] += red[t + s][0];
            red[t][1] += red[t + s][1];
            red[t][2] += red[t + s][2];
            red[t][3] += red[t + s][3];
        }
        __syncthreads();
    }
    if (t == 0) {
        float la = red[0][0] / fmaxf(red[0][1], EPSF);
        float lt = red[0][2] / fmaxf(red[0][3], EPSF);
        atomicAdd(&out[0], la * (1.0f / NB));
        atomicAdd(&out[1], lt * (1.0f / NB));
    }
}

// ---------------------------------------------------------------------------
// dist + clash via V_WMMA_F32_16X16X4_F32 squared-distance tiles
//   A row m  = (x_m, y_m, z_m, 1)
//   B col n  = (-2x_n, -2y_n, -2z_n, sq_n)
//   C[m][n]  = sq_m
//   D = A*B + C = sq_m + sq_n - 2 x_m . x_n
// grid: (16 j-tile chunks of 4, 64 i-tiles, 16 batches), block = 1 wave32
// ---------------------------------------------------------------------------
__global__ __launch_bounds__(32)
void k_dist(const float4* __restrict__ pc, const float4* __restrict__ tc,
            const int* __restrict__ nvalid, float* __restrict__ acc) {
    const int b  = blockIdx.z;
    const int nv = nvalid[b];
    const int i0 = blockIdx.y * 16;
    if (i0 >= nv) return;                 // uniform: tile fully masked out

    const int lane = threadIdx.x;
    const int half = lane >> 4;           // 0: lanes 0-15, 1: lanes 16-31
    const int l16  = lane & 15;
    const int base = half ? 8 : 0;

    const float4* P = pc + b * LSEQ;
    const float4* T = tc + b * LSEQ;

    // A fragments (16x4 f32): lanes 0-15 -> K={0,1}=(x,y); lanes 16-31 -> K={2,3}=(z,1)
    float4 arP = P[i0 + l16];
    float4 arT = T[i0 + l16];
    v2f aP, aT;
    aP[0] = half ? arP.z : arP.x;  aP[1] = half ? 1.0f : arP.y;
    aT[0] = half ? arT.z : arT.x;  aT[1] = half ? 1.0f : arT.y;

    // C init: C vgpr r holds sq of row i0 + r + 8*half (broadcast across lanes)
    v8f cP, cT;
    #pragma unroll
    for (int r = 0; r < 8; ++r) {
        cP[r] = __shfl(arP.w, base + r, 32);
        cT[r] = __shfl(arT.w, base + r, 32);
    }

    float accD = 0.f, accC = 0.f;
    const int tj0 = blockIdx.x * 4;
    for (int tj = tj0; tj < tj0 + 4; ++tj) {
        const int j0 = tj * 16;
        if (j0 >= nv) break;              // uniform

        float4 bcP = P[j0 + l16];
        float4 bcT = T[j0 + l16];
        // B fragments (4x16 f32): lanes 0-15 -> K={0,1}; lanes 16-31 -> K={2,3}
        v2f bP, bT;
        bP[0] = half ? (-2.f * bcP.z) : (-2.f * bcP.x);
        bP[1] = half ? bcP.w          : (-2.f * bcP.y);
        bT[0] = half ? (-2.f * bcT.z) : (-2.f * bcT.x);
        bT[1] = half ? bcT.w          : (-2.f * bcT.y);

        v8f dP = __builtin_amdgcn_wmma_f32_16x16x4_f32(
            false, aP, false, bP, (short)0, cP, false, false);
        v8f dT = __builtin_amdgcn_wmma_f32_16x16x4_f32(
            false, aT, false, bT, (short)0, cT, false, false);

        const int col = j0 + l16;
        #pragma unroll
        for (int r = 0; r < 8; ++r) {
            const int row = i0 + base + r;
            float w = (row < nv && col < nv) ? 1.0f : 0.0f;
            float dp2 = dP[r];
            float dt2 = dT[r];
            float d1 = (dp2 > 0.f) ? fast_sqrt(dp2) : 0.f;
            float d2 = (dt2 > 0.f) ? fast_sqrt(dt2) : 0.f;
            float diff = d1 - d2;
            accD += w * diff * diff;
            float cl = (d1 < 3.8f && d1 > 2.0f) ? (3.8f - d1) * (3.8f - d1) : 0.f;
            accC += w * cl;
        }
    }

    // wave32 reduction
    #pragma unroll
    for (int off = 16; off > 0; off >>= 1) {
        accD += __shfl_down(accD, off, 32);
        accC += __shfl_down(accC, off, 32);
    }
    if (lane == 0) {
        atomicAdd(&acc[b], accD);
        atomicAdd(&acc[NB + b], accC);
    }
}

// ---------------------------------------------------------------------------
// finalize: dist_loss = mean_b(sum_b / (nv^2 + 1e-8)), same for clash
// ---------------------------------------------------------------------------
__global__ void k_fin(const float* __restrict__ acc, const int* __restrict__ nvalid,
                      float* __restrict__ out) {
    int t = threadIdx.x;
    if (t < NB) {
        float nv = (float)nvalid[t];
        float denom = nv * nv + 1e-8f;
        atomicAdd(&out[2], (acc[t]      / denom) * (1.0f / NB));
        atomicAdd(&out[3], (acc[NB + t] / denom) * (1.0f / NB));
    }
}

// ---------------------------------------------------------------------------
extern "C" void kernel_launch(void* const* d_in, const int* in_sizes, int n_in,
                              void* d_out, int out_size, void* d_ws, size_t ws_size,
                              hipStream_t stream) {
    const float* pred   = (const float*)d_in[0];
    const float* target = (const float*)d_in[1];
    const int*   gmask  = (const int*)d_in[2];
    float* out = (float*)d_out;

    char* ws = (char*)d_ws;
    float4* pc = (float4*)ws;                         // 16*1024 float4 = 256 KB
    float4* tc = pc + NB * LSEQ;                      // 256 KB
    int* nvalid = (int*)(tc + NB * LSEQ);             // 16 ints
    float* acc = (float*)(nvalid + NB);               // 32 floats

    k_init<<<1, 64, 0, stream>>>(out, acc);
    k_compact<<<NB, LSEQ, 0, stream>>>(pred, target, gmask, pc, tc, nvalid);
    k_angle<<<NB, 256, 0, stream>>>(pc, tc, nvalid, out);
    k_dist<<<dim3(16, 64, NB), 32, 0, stream>>>(pc, tc, nvalid, acc);
    k_fin<<<1, 32, 0, stream>>>(acc, nvalid, out);
}